// WaterRDFMAE_51213190037900
// MI455X (gfx1250) — compile-verified
//
#include <hip/hip_runtime.h>
#include <stdint.h>

// ---------------------------------------------------------------------------
// Water RDF + MAE for MI455X (gfx1250).
//   in[0] stacked_radii f32 [64][8][192][3]
//   in[1] lattices      f32 [3]
//   in[2..4] gt_oo/hh/ho f32 [500]
//   in[5] o_idx i32 [64], in[6] h_idx i32 [128]
//   out: [8][1500] rdf rows (oo|hh|ho) + [8] max_maes  (12008 f32)
//   ws : 8*1500 u32 count accumulators (48 KB), zeroed every call.
//
// CDNA5 features used:
//   - global_load_async_to_lds_b128 (async-tensor path, ASYNCcnt) to stage
//     each frame's 2304 B of positions into LDS
//   - s_wait_asynccnt 0 dependency wait
//   - ds_add_u32 no-return LDS atomics for privatized histograms
// ---------------------------------------------------------------------------

#define TFR   64
#define RREP  8
#define NA    192
#define NO    64
#define NH    128
#define NBINS 500
#define CHUNK (NA * 3)   // 576 floats per (frame, replica)

static constexpr float B0F   = 1e-6f;
static constexpr float STEPF = (float)((10.0 - 1e-6) / 500.0);
static constexpr float ELAST = 10.0f;   // linspace endpoint is exact

#if defined(__has_builtin)
#if __has_builtin(__builtin_amdgcn_s_wait_asynccnt)
#define HAVE_WAIT_ASYNC 1
#endif
#endif

__device__ __forceinline__ float edgef(int k) {
  return fmaf((float)k, STEPF, B0F);   // BINS[k]
}

// np.histogram semantics: idx = searchsorted(bins, d, 'right') - 1,
// with d == bins[-1] mapped to the last bin; out-of-range -> -1.
__device__ __forceinline__ int bin_index(float d) {
  if (!(d >= B0F)) return -1;          // below bins[0] (covers self-pairs d=0)
  if (d == ELAST) return NBINS - 1;
  int k = (int)((d - B0F) * (1.0f / STEPF));
  if (k > NBINS) k = NBINS;
  if (d < edgef(k))            --k;    // fp one-off correction
  else if (d >= edgef(k + 1))  ++k;
  return (k >= 0 && k < NBINS) ? k : -1;
}

__device__ __forceinline__ float min_image(float d, float L, float hL) {
  d = fabsf(d);
  return (d > hL) ? (d - L) : d;
}

__device__ __forceinline__ float wrap_coord(float x, float L, float invL) {
  float u = x * invL;
  u -= floorf(u);                      // jnp.mod(x/L, 1.0)
  return u * L;
}

__global__ void zero_acc_kernel(uint32_t* __restrict__ acc, int n) {
  int i = blockIdx.x * blockDim.x + threadIdx.x;
  if (i < n) acc[i] = 0u;
}

__global__ __launch_bounds__(256)
void rdf_hist_kernel(const float* __restrict__ pos,
                     const float* __restrict__ lat,
                     const int*   __restrict__ o_idx,
                     const int*   __restrict__ h_idx,
                     uint32_t*    __restrict__ acc) {
  __shared__ __align__(16) float stage[CHUNK];           // raw frame positions
  __shared__ float sox[NO], soy[NO], soz[NO];            // wrapped O (SoA)
  __shared__ float shx[NH], shy[NH], shz[NH];            // wrapped H (SoA)
  __shared__ uint32_t hist[3 * NBINS];                   // per-WG histograms

  const int tid = threadIdx.x;
  const int bid = blockIdx.x;                            // t*R + r
  const int r   = bid & (RREP - 1);
  const float* chunk = pos + (size_t)bid * CHUNK;

  // ---- async global->LDS stage of this frame's 2304 B (144 x b128 lanes) --
  // vDST = 32-bit LDS byte offset (generic->AS3 is a truncate),
  // vADDR = 64-bit global address, saddr = off.
  if (tid < CHUNK / 4) {
    const uint32_t lds_off = (uint32_t)(uintptr_t)(stage + tid * 4);
    const uint64_t gaddr   = (uint64_t)(uintptr_t)(chunk + tid * 4);
    asm volatile("global_load_async_to_lds_b128 %0, %1, off"
                 :: "v"(lds_off), "v"(gaddr) : "memory");
  }

  // zero histograms while the async copy is in flight
  for (int b = tid; b < 3 * NBINS; b += 256) hist[b] = 0u;

  const float Lx = lat[0], Ly = lat[1], Lz = lat[2];
  const float iLx = 1.0f / Lx, iLy = 1.0f / Ly, iLz = 1.0f / Lz;
  const float hLx = 0.5f * Lx, hLy = 0.5f * Ly, hLz = 0.5f * Lz;

#ifdef HAVE_WAIT_ASYNC
  __builtin_amdgcn_s_wait_asynccnt(0);
#else
  asm volatile("s_wait_asynccnt 0" ::: "memory");
#endif
  __syncthreads();

  // ---- PBC-wrap and scatter to SoA O/H arrays -----------------------------
  if (tid < NO) {
    const int a = o_idx[tid];
    sox[tid] = wrap_coord(stage[a * 3 + 0], Lx, iLx);
    soy[tid] = wrap_coord(stage[a * 3 + 1], Ly, iLy);
    soz[tid] = wrap_coord(stage[a * 3 + 2], Lz, iLz);
  } else if (tid < NO + NH) {
    const int k = tid - NO;
    const int a = h_idx[k];
    shx[k] = wrap_coord(stage[a * 3 + 0], Lx, iLx);
    shy[k] = wrap_coord(stage[a * 3 + 1], Ly, iLy);
    shz[k] = wrap_coord(stage[a * 3 + 2], Lz, iLz);
  }
  __syncthreads();

  // ---- O-O : 64x64 = 4096 pairs, 16 per thread; j fixed per thread --------
  {
    const int j = tid & 63;
    const float xj = sox[j], yj = soy[j], zj = soz[j];
    const int i0 = tid >> 6;                 // wave-uniform
#pragma unroll
    for (int k = 0; k < 16; ++k) {
      const int i = i0 + (k << 2);           // broadcast LDS read
      const float dx = min_image(sox[i] - xj, Lx, hLx);
      const float dy = min_image(soy[i] - yj, Ly, hLy);
      const float dz = min_image(soz[i] - zj, Lz, hLz);
      const float d  = sqrtf(fmaf(dx, dx, fmaf(dy, dy, dz * dz)));
      const int  b   = bin_index(d);
      if (b >= 0) atomicAdd(&hist[b], 1u);   // ds_add_u32 (no return)
    }
  }

  // ---- H-H : 128x128 = 16384 pairs, 64 per thread -------------------------
  {
    const int j = tid & 127;
    const float xj = shx[j], yj = shy[j], zj = shz[j];
    const int i0 = tid >> 7;                 // wave-uniform
#pragma unroll 8
    for (int k = 0; k < 64; ++k) {
      const int i = i0 + (k << 1);
      const float dx = min_image(shx[i] - xj, Lx, hLx);
      const float dy = min_image(shy[i] - yj, Ly, hLy);
      const float dz = min_image(shz[i] - zj, Lz, hLz);
      const float d  = sqrtf(fmaf(dx, dx, fmaf(dy, dy, dz * dz)));
      const int  b   = bin_index(d);
      if (b >= 0) atomicAdd(&hist[NBINS + b], 1u);
    }
  }

  // ---- H-O : 128x64 = 8192 pairs, 32 per thread (i over H, j over O) ------
  {
    const int j = tid & 63;
    const float xj = sox[j], yj = soy[j], zj = soz[j];
    const int i0 = tid >> 6;                 // wave-uniform
#pragma unroll 8
    for (int k = 0; k < 32; ++k) {
      const int i = i0 + (k << 2);
      const float dx = min_image(shx[i] - xj, Lx, hLx);
      const float dy = min_image(shy[i] - yj, Ly, hLy);
      const float dz = min_image(shz[i] - zj, Lz, hLz);
      const float d  = sqrtf(fmaf(dx, dx, fmaf(dy, dy, dz * dz)));
      const int  b   = bin_index(d);
      if (b >= 0) atomicAdd(&hist[2 * NBINS + b], 1u);
    }
  }

  __syncthreads();

  // ---- flush WG histogram into per-replica global accumulator -------------
  uint32_t* row = acc + (size_t)r * (3 * NBINS);
  for (int b = tid; b < 3 * NBINS; b += 256) {
    const uint32_t c = hist[b];
    if (c) atomicAdd(&row[b], c);            // global_atomic_add_u32 (L2)
  }
}

__global__ __launch_bounds__(256)
void rdf_finalize_kernel(const uint32_t* __restrict__ acc,
                         const float* __restrict__ lat,
                         const float* __restrict__ gt_oo,
                         const float* __restrict__ gt_hh,
                         const float* __restrict__ gt_ho,
                         float* __restrict__ out) {
  const int r = blockIdx.x;
  const int tid = threadIdx.x;
  __shared__ float red0[256], red1[256], red2[256];

  const float V = lat[0] * lat[1] * lat[2];
  const float c43pi = 4.18879020478639098461f;          // 4/3 * pi
  float s0 = 0.0f, s1 = 0.0f, s2 = 0.0f;

  for (int b = tid; b < NBINS; b += 256) {
    const float e0 = edgef(b), e1 = edgef(b + 1);
    const float shell = c43pi * (e1 * e1 * e1 - e0 * e0 * e0);
    // rdf = cnt / (T * Z),  Z = (n_pairs/V) * shell
    const float inv = V / ((float)TFR * shell);
    const float roo = (float)acc[r * 1500 + b]             * (inv / 4096.0f);
    const float rhh = (float)acc[r * 1500 + NBINS + b]     * (inv / 16384.0f);
    const float rho = (float)acc[r * 1500 + 2 * NBINS + b] * (inv / 8192.0f);
    out[r * 1500 + b]             = roo;
    out[r * 1500 + NBINS + b]     = rhh;
    out[r * 1500 + 2 * NBINS + b] = rho;
    s0 += fabsf(roo - gt_oo[b]);
    s1 += fabsf(rhh - gt_hh[b]);
    s2 += fabsf(rho - gt_ho[b]);
  }

  red0[tid] = s0; red1[tid] = s1; red2[tid] = s2;
  __syncthreads();
  for (int s = 128; s > 0; s >>= 1) {
    if (tid < s) {
      red0[tid] += red0[tid + s];
      red1[tid] += red1[tid + s];
      red2[tid] += red2[tid + s];
    }
    __syncthreads();
  }
  if (tid == 0) {
    const float m0 = 10.0f * red0[0] / 500.0f;   // XLIM * mean(|.|)
    const float m1 = 10.0f * red1[0] / 500.0f;
    const float m2 = 10.0f * red2[0] / 500.0f;
    out[RREP * 1500 + r] = fmaxf(m0, fmaxf(m1, m2));   // out[12000 + r]
  }
}

extern "C" void kernel_launch(void* const* d_in, const int* in_sizes, int n_in,
                              void* d_out, int out_size, void* d_ws, size_t ws_size,
                              hipStream_t stream) {
  const float* pos   = (const float*)d_in[0];
  const float* lat   = (const float*)d_in[1];
  const float* gt_oo = (const float*)d_in[2];
  const float* gt_hh = (const float*)d_in[3];
  const float* gt_ho = (const float*)d_in[4];
  const int*   o_idx = (const int*)d_in[5];
  const int*   h_idx = (const int*)d_in[6];
  float*       out   = (float*)d_out;
  uint32_t*    acc   = (uint32_t*)d_ws;          // 8*1500 u32 = 48 KB

  const int nacc = RREP * 3 * NBINS;             // 12000
  zero_acc_kernel<<<(nacc + 255) / 256, 256, 0, stream>>>(acc, nacc);
  rdf_hist_kernel<<<TFR * RREP, 256, 0, stream>>>(pos, lat, o_idx, h_idx, acc);
  rdf_finalize_kernel<<<RREP, 256, 0, stream>>>(acc, lat, gt_oo, gt_hh, gt_ho, out);
}